// Select_Topk_88175678587540
// MI455X (gfx1250) — compile-verified
//
#include <hip/hip_runtime.h>
#include <hip/hip_bf16.h>
#include <cstdint>

// Problem constants (from reference): B=16, N=200, C=768, CE=128, K=100
#define BB 16
#define NN 200
#define CC 768
#define CE 128
#define KK 100

// d_out layout (tuple concatenated flat, f32):
//   nodes: [16,100,768]      -> offset 0,        1,228,800 elems
//   mask : [16,100]          -> offset 1228800,      1,600 elems
//   edges: [16,100,100,128]  -> offset 1230400, 20,480,000 elems

// ---------------------------------------------------------------------------
// Kernel 1: exact top-k (k smallest, ties -> lower index), output indices
// sorted ascending by index into workspace; also writes the ones-mask.
// One block per batch; N=200 so O(N^2) rank is trivial.
// ---------------------------------------------------------------------------
__global__ __launch_bounds__(256)
void topk_select_kernel(const float* __restrict__ scores,
                        int* __restrict__ idx_ws,
                        float* __restrict__ mask_out) {
    const int b = blockIdx.x;
    const int t = threadIdx.x;

    __shared__ float s[NN];
    __shared__ int   sel[NN];

    if (t < NN) s[t] = scores[b * NN + t];
    __syncthreads();

    if (t < NN) {
        const float v = s[t];
        int rank = 0;
        #pragma unroll 8
        for (int j = 0; j < NN; ++j) {
            const float u = s[j];
            // lexicographic (value, index): matches top_k(-scores) stability
            rank += (u < v) || (u == v && j < t);
        }
        sel[t] = (rank < KK) ? 1 : 0;
    }
    __syncthreads();

    if (t < NN && sel[t]) {
        // position in ascending-index compaction = #selected with smaller index
        int pos = 0;
        for (int j = 0; j < t; ++j) pos += sel[j];
        idx_ws[b * KK + pos] = t;
    }
    if (t < KK) mask_out[b * KK + t] = 1.0f;
}

// ---------------------------------------------------------------------------
// Kernel 2: node gather, coalesced float4.  out[b,r,:] = v_feat[b, idx[b,r], :]
// 16*100 rows * 192 float4 per row = 307,200 float4 transfers.
// ---------------------------------------------------------------------------
__global__ __launch_bounds__(256)
void node_gather_kernel(const float* __restrict__ v_feat,
                        const int* __restrict__ idx_ws,
                        float* __restrict__ nodes_out) {
    const int tid = blockIdx.x * blockDim.x + threadIdx.x;
    const int TOT = BB * KK * (CC / 4);           // 307200
    if (tid >= TOT) return;
    const int c4  = tid % (CC / 4);
    const int rem = tid / (CC / 4);
    const int r   = rem % KK;
    const int b   = rem / KK;
    const int src = idx_ws[b * KK + r];
    const float4* in = reinterpret_cast<const float4*>(v_feat + ((size_t)b * NN + src) * CC);
    float4*       o  = reinterpret_cast<float4*>(nodes_out + ((size_t)b * KK + r) * CC);
    o[c4] = in[c4];
}

// ---------------------------------------------------------------------------
// gfx1250 async LDS DMA helpers (ASYNCcnt path; data never enters VGPRs).
// ---------------------------------------------------------------------------
__device__ __forceinline__ void async_load_b128(uint32_t lds, uint64_t gaddr) {
    asm volatile("global_load_async_to_lds_b128 %0, %1, off"
                 :: "v"(lds), "v"(gaddr) : "memory");
}
__device__ __forceinline__ void async_store_b128(uint64_t gaddr, uint32_t lds) {
    asm volatile("global_store_async_from_lds_b128 %0, %1, off"
                 :: "v"(gaddr), "v"(lds) : "memory");
}
__device__ __forceinline__ void wait_async0() {
    asm volatile("s_wait_asynccnt 0" ::: "memory");
}

// ---------------------------------------------------------------------------
// Kernel 3: edge gather via async LDS DMA, 4-deep batched double buffering.
// One edge row = CE*4 = 512 B = wave32 x b128 -> one async op per row.
// Per wave: 2 banks x 4 slots x 512 B. Steady state between waits:
//   4 stores (bank cur) + 4 loads (bank oth) in flight = 4 KB/wave.
// 10,000 waves * 4 KB = 40 MB outstanding -> covers the ~23 MB
// latency-bandwidth product of 23.3 TB/s HBM.
// Every s_wait_asynccnt 0 retires all prior ops, so loads never overwrite a
// slot before its store completed, and stores never read a slot before its
// load completed.
// ---------------------------------------------------------------------------
#define EBATCH 4

__global__ __launch_bounds__(256)
void edge_gather_async_kernel(const float* __restrict__ v2v,
                              const int* __restrict__ idx_ws,
                              float* __restrict__ edges_out) {
    __shared__ __align__(16) char smem[8 * 2 * EBATCH * 512];   // 32 KB/block

    const int lane = threadIdx.x & 31;
    const int wib  = threadIdx.x >> 5;
    const int gwave  = (blockIdx.x * blockDim.x + threadIdx.x) >> 5;
    const int nwaves = (gridDim.x * blockDim.x) >> 5;
    const int R = BB * KK * KK;                        // 160000 rows

    // Low 32 bits of a flat shared-space pointer == LDS byte offset (ISA 10.2).
    const uint32_t ldsA = (uint32_t)(uintptr_t)(smem + wib * (2 * EBATCH * 512))
                          + (uint32_t)(lane * 16);
    const uint32_t ldsB = ldsA + EBATCH * 512;

    int r = gwave;
    if (r >= R) return;                                // uniform per wave

    auto src_addr = [&](int t) -> uint64_t {
        const int b   = t / (KK * KK);
        const int rem = t % (KK * KK);
        const int rr  = rem / KK;
        const int rc  = rem % KK;
        const int i0  = idx_ws[b * KK + rr];
        const int i1  = idx_ws[b * KK + rc];
        const float* p = v2v + (((size_t)b * NN + i0) * NN + i1) * CE;
        return (uint64_t)(uintptr_t)p + (uint64_t)(lane * 16);
    };
    auto dst_addr = [&](int t) -> uint64_t {
        return (uint64_t)(uintptr_t)(edges_out + (size_t)t * CE) + (uint64_t)(lane * 16);
    };

    uint32_t cur = ldsA, oth = ldsB;

    // Prologue: first batch of rows in flight.
    {
        int t = r;
        #pragma unroll
        for (int i = 0; i < EBATCH; ++i) {
            if (t < R) async_load_b128(cur + i * 512, src_addr(t));
            t += nwaves;
        }
    }

    for (;;) {
        const int nbase = r + EBATCH * nwaves;
        // Retire all outstanding: current-batch loads and previous-batch stores.
        wait_async0();
        // Next batch of loads into the other bank (overlaps with stores below).
        {
            int t = nbase;
            #pragma unroll
            for (int i = 0; i < EBATCH; ++i) {
                if (t < R) async_load_b128(oth + i * 512, src_addr(t));
                t += nwaves;
            }
        }
        // Stores for the current batch straight out of LDS.
        {
            int t = r;
            #pragma unroll
            for (int i = 0; i < EBATCH; ++i) {
                if (t < R) async_store_b128(dst_addr(t), cur + i * 512);
                t += nwaves;
            }
        }
        if (nbase >= R) break;
        r = nbase;
        const uint32_t tmp = cur; cur = oth; oth = tmp;
    }
    // Drain before S_ENDPGM (endpgm also waits; this makes it explicit).
    wait_async0();
}

// ---------------------------------------------------------------------------
extern "C" void kernel_launch(void* const* d_in, const int* in_sizes, int n_in,
                              void* d_out, int out_size, void* d_ws, size_t ws_size,
                              hipStream_t stream) {
    const float* obj_scores = (const float*)d_in[0];   // [16,200]
    const float* v_feat     = (const float*)d_in[1];   // [16,200,768]
    const float* v2v_edge   = (const float*)d_in[2];   // [16,200,200,128]

    float* out = (float*)d_out;
    float* nodes_out = out;                            // 1,228,800
    float* mask_out  = out + (size_t)BB * KK * CC;     // 1,600
    float* edges_out = mask_out + (size_t)BB * KK;     // 20,480,000

    int* idx_ws = (int*)d_ws;                          // [16,100] ints (6.4 KB)

    // 1) top-k indices + mask (16 blocks, one per batch)
    topk_select_kernel<<<BB, 256, 0, stream>>>(obj_scores, idx_ws, mask_out);

    // 2) node gather: 307200 float4 / 256 = 1200 blocks
    node_gather_kernel<<<(BB * KK * (CC / 4) + 255) / 256, 256, 0, stream>>>(
        v_feat, idx_ws, nodes_out);

    // 3) edge gather: 160000 rows over 10,000 waves (1250 blocks x 8 waves),
    //    16 rows per wave = 4 batches of EBATCH=4.
    edge_gather_async_kernel<<<1250, 256, 0, stream>>>(v2v_edge, idx_ws, edges_out);
}